// FocalLossWithSmoothing_9826885174161
// MI455X (gfx1250) — compile-verified
//
#include <hip/hip_runtime.h>

#ifndef __has_builtin
#define __has_builtin(x) 0
#endif

#if __has_builtin(__builtin_amdgcn_global_load_async_to_lds_b128)
#define ATHENA_ASYNC_LDS 1
#else
#define ATHENA_ASYNC_LDS 0
#endif

typedef int athena_v4i __attribute__((ext_vector_type(4)));
typedef __attribute__((address_space(1))) athena_v4i* athena_gv4i_p;  // global
typedef __attribute__((address_space(3))) athena_v4i* athena_lv4i_p;  // LDS

namespace {
constexpr int   kC     = 32000;        // classes per row
constexpr int   kC4    = kC / 4;       // float4 count per row
constexpr int   kBlock = 1024;         // 32 waves (wave32)
constexpr int   kWaves = kBlock / 32;
constexpr float kPos   = 1.0f - 0.1f;              // 1 - LABEL_SMOOTHING
constexpr float kNeg   = 0.1f / (float)(kC - 1);   // LABEL_SMOOTHING / (C-1)
}

static __device__ __forceinline__ void wait_async_zero() {
#if __has_builtin(__builtin_amdgcn_s_wait_asynccnt)
  __builtin_amdgcn_s_wait_asynccnt(0);
#else
  asm volatile("s_wait_asynccnt 0" ::: "memory");
#endif
}

static __device__ __forceinline__ float wave_max(float v) {
  for (int off = 16; off; off >>= 1) v = fmaxf(v, __shfl_xor(v, off, 32));
  return v;
}
static __device__ __forceinline__ float wave_sum(float v) {
  for (int off = 16; off; off >>= 1) v += __shfl_xor(v, off, 32);
  return v;
}

// One workgroup per row. Row (128 KB) is DMA'd into LDS once via the gfx1250
// async global->LDS path (ASYNCcnt + s_wait_asynccnt). Then:
//   pass A: row max (LDS read)
//   pass B: e = exp(x - max) written back over the row in LDS; Z = sum(e)
//   pass C: focal negative terms from cached e (one v_log per element)
// Trans budget: 1 exp + 1 log per element; HBM touched exactly once.
__global__ __launch_bounds__(kBlock, 2) void focal_rows_kernel(
    const float* __restrict__ X, const int* __restrict__ T,
    float* __restrict__ row_loss) {
  extern __shared__ float lds[];        // kC row floats + kWaves reduction slots
  float* red = lds + kC;

  const int tid  = (int)threadIdx.x;
  const int lane = tid & 31;
  const int wave = tid >> 5;
  const int row  = (int)blockIdx.x;
  const float* __restrict__ x = X + (size_t)row * (size_t)kC;

  // ---- Stage 1: global -> LDS, async DMA path ----
#if ATHENA_ASYNC_LDS
  for (int i = tid; i < kC4; i += kBlock) {
    __builtin_amdgcn_global_load_async_to_lds_b128(
        (athena_gv4i_p)(x + 4 * i),
        (athena_lv4i_p)(lds + 4 * i),
        0, 0);
  }
  wait_async_zero();                    // this wave's async loads done
#else
  for (int i = tid; i < kC4; i += kBlock) {
    const float4 v = reinterpret_cast<const float4*>(x)[i];
    reinterpret_cast<float4*>(lds)[i] = v;
  }
#endif
  __syncthreads();                      // all waves' LDS data visible

  const int tgt = T[row];

  // ---- Stage 2: row max ----
  float m = -__builtin_inff();
  for (int i = tid; i < kC4; i += kBlock) {
    const float4 v = reinterpret_cast<const float4*>(lds)[i];
    m = fmaxf(m, fmaxf(fmaxf(v.x, v.y), fmaxf(v.z, v.w)));
  }
  m = wave_max(m);
  if (lane == 0) red[wave] = m;
  __syncthreads();
  const float bm = wave_max(red[lane]); // every thread gets block max
  const float xt = lds[tgt];            // original target logit (pre-overwrite)
  __syncthreads();                      // red[] + xt reads done before reuse

  // ---- Stage 3: e = exp(x - max) cached back into LDS; Z = sum(e) ----
  // Each thread overwrites exactly the indices it reads -> no cross-thread
  // hazard; the barrier above protects the cross-thread xt read.
  float se = 0.0f;
  for (int i = tid; i < kC4; i += kBlock) {
    float4 v = reinterpret_cast<const float4*>(lds)[i];
    v.x = __expf(v.x - bm);
    v.y = __expf(v.y - bm);
    v.z = __expf(v.z - bm);
    v.w = __expf(v.w - bm);
    reinterpret_cast<float4*>(lds)[i] = v;
    se += v.x + v.y + v.z + v.w;
  }
  se = wave_sum(se);
  if (lane == 0) red[wave] = se;
  __syncthreads();
  const float Z    = wave_sum(red[lane]);
  __syncthreads();                      // Z reads done; e-writes visible
  const float invZ = 1.0f / Z;
  const float logZ = __logf(Z);

  // ---- Stage 4: negative-class focal terms (p = e * invZ), excl. target ----
  float acc = 0.0f;
  for (int i = tid; i < kC4; i += kBlock) {
    const float4 v = reinterpret_cast<const float4*>(lds)[i];
    const int j = 4 * i;
    {
      const float p = v.x * invZ;
      const float t = p * p * __logf(1.0f - p);
      acc += (j + 0 == tgt) ? 0.0f : t;
    }
    {
      const float p = v.y * invZ;
      const float t = p * p * __logf(1.0f - p);
      acc += (j + 1 == tgt) ? 0.0f : t;
    }
    {
      const float p = v.z * invZ;
      const float t = p * p * __logf(1.0f - p);
      acc += (j + 2 == tgt) ? 0.0f : t;
    }
    {
      const float p = v.w * invZ;
      const float t = p * p * __logf(1.0f - p);
      acc += (j + 3 == tgt) ? 0.0f : t;
    }
  }
  acc = wave_sum(acc);
  if (lane == 0) red[wave] = acc;
  __syncthreads();
  const float negsum = wave_sum(red[lane]) * kNeg;

  // ---- Stage 5: positive-class term and per-row loss ----
  if (tid == 0) {
    const float et    = lds[tgt];                // exp(xt - bm) cached in LDS
    const float pt    = et * invZ;
    const float logpt = (xt - bm) - logZ;        // exact log softmax at target
    const float om    = 1.0f - pt;
    const float posl  = kPos * om * om * logpt;  // gamma = 2
    row_loss[row] = -(posl + negsum);
  }
}

// Single-block deterministic mean over row losses.
__global__ __launch_bounds__(1024) void focal_mean_kernel(
    const float* __restrict__ rl, float* __restrict__ out, int n) {
  __shared__ float red[32];
  const int tid  = (int)threadIdx.x;
  const int lane = tid & 31;
  const int wave = tid >> 5;
  float s = 0.0f;
  for (int i = tid; i < n; i += 1024) s += rl[i];
  s = wave_sum(s);
  if (lane == 0) red[wave] = s;
  __syncthreads();
  if (wave == 0) {
    float t = wave_sum(red[lane]);
    if (lane == 0) out[0] = t / (float)n;
  }
}

extern "C" void kernel_launch(void* const* d_in, const int* in_sizes, int n_in,
                              void* d_out, int out_size, void* d_ws, size_t ws_size,
                              hipStream_t stream) {
  (void)n_in; (void)out_size; (void)ws_size;
  const float* X = (const float*)d_in[0];
  const int*   T = (const int*)d_in[1];
  float* out     = (float*)d_out;
  float* rowloss = (float*)d_ws;         // N floats of scratch (16 KB)

  const int N = in_sizes[1];             // 4096 rows (one block each)

  const size_t shmem = (size_t)(kC + kWaves) * sizeof(float);  // ~128.1 KB
  // Opt in to >64KB dynamic LDS (gfx1250 WGP has 320 KB).
  (void)hipFuncSetAttribute(reinterpret_cast<const void*>(focal_rows_kernel),
                            hipFuncAttributeMaxDynamicSharedMemorySize,
                            (int)shmem);

  focal_rows_kernel<<<N, kBlock, shmem, stream>>>(X, T, rowloss);
  focal_mean_kernel<<<1, 1024, 0, stream>>>(rowloss, out, N);
}